// ModifiedMarginRankingLoss_69355131896106
// MI455X (gfx1250) — compile-verified
//
#include <hip/hip_runtime.h>
#include <hip/hip_bf16.h>

typedef __attribute__((ext_vector_type(2))) float v2f;
typedef __attribute__((ext_vector_type(8))) float v8f;

#define MARGIN_F 0.1f
#define TILE 16
#define CHUNK 32            // tiles along j handled per wave
#define WAVES_PER_BLOCK 8
#define BLOCK_THREADS 256   // 8 wave32's per block

// Kernel 1: gather masked subset, transform, pad to multiple of 16.
// g_l[k] = (float)labels[mask[k]], g_u[k] = MARGIN*l - s.
// Padding uses (l=-1e30, u=+1e30) which makes every padded pair's hinge 0:
//   pad as "low label" side  -> relu(u_real - 1e30) = 0
//   pad as "high label" side -> impossible (l_pad smallest)
//   pad-pad                  -> ld = 0 -> 0
__global__ void mmrl_gather_kernel(const float* __restrict__ scores,
                                   const int* __restrict__ labels,
                                   const int* __restrict__ mask,
                                   float* __restrict__ g_l,
                                   float* __restrict__ g_u,
                                   int M, int M16) {
    int k = blockIdx.x * blockDim.x + threadIdx.x;
    if (k >= M16) return;
    if (k < M) {
        int idx = mask[k];
        float l = (float)labels[idx];
        float s = scores[idx];
        g_l[k] = l;
        g_u[k] = MARGIN_F * l - s;
    } else {
        g_l[k] = -1e30f;
        g_u[k] =  1e30f;
    }
}

// Kernel 2: one wave per (tile-row, j-chunk). For each 16x16 tile, two
// V_WMMA_F32_16X16X4_F32 ops produce the outer-difference tiles
//   ld[i][j] = l_i - l_j,  du[i][j] = u_i - u_j
// via A=[x_i, 1] (16x4), B=[1; -x_j] (4x16). VALU applies
//   h = relu(sign(ld) * du)
// per element and accumulates. Per-wave partial written to workspace
// (no float atomics -> deterministic).
__global__ void mmrl_pair_tile_kernel(const float* __restrict__ g_l,
                                      const float* __restrict__ g_u,
                                      float* __restrict__ partials,
                                      int T, int nchunk) {
    const int lane = threadIdx.x & 31;
    const int w = blockIdx.x * WAVES_PER_BLOCK + (threadIdx.x >> 5);

    float acc = 0.0f;
    const int nwork = T * nchunk;
    if (w < nwork) {
        const int ti = w / nchunk;
        const int chunk = w - ti * nchunk;
        const int tj_begin = chunk * CHUNK;
        int tj_end = tj_begin + CHUNK;
        if (tj_end > T) tj_end = T;

        const int  r  = lane & 15;
        const bool lo = lane < 16;   // lanes 0-15 carry K=0/1; lanes 16-31 (K=2/3) zeroed

        const float li = g_l[ti * TILE + r];
        const float ui = g_u[ti * TILE + r];

        // A (16x4, MxK): col K=0 = x_i, col K=1 = 1, K=2/3 = 0
        v2f a_l, a_u;
        a_l.x = lo ? li   : 0.0f;  a_l.y = lo ? 1.0f : 0.0f;
        a_u.x = lo ? ui   : 0.0f;  a_u.y = lo ? 1.0f : 0.0f;

        for (int tj = tj_begin; tj < tj_end; ++tj) {
            const float lj = g_l[tj * TILE + r];
            const float uj = g_u[tj * TILE + r];

            // B (4x16, KxN): row K=0 = 1, row K=1 = -x_j, K=2/3 = 0
            v2f b_l, b_u;
            b_l.x = lo ? 1.0f : 0.0f;  b_l.y = lo ? -lj : 0.0f;
            b_u.x = lo ? 1.0f : 0.0f;  b_u.y = lo ? -uj : 0.0f;

            v8f c = {};
            v8f dl = __builtin_amdgcn_wmma_f32_16x16x4_f32(
                false, a_l, false, b_l, (short)0, c, false, false);
            v8f du = __builtin_amdgcn_wmma_f32_16x16x4_f32(
                false, a_u, false, b_u, (short)0, c, false, false);

#pragma unroll
            for (int e = 0; e < 8; ++e) {
                float ld = dl[e];
                float dd = du[e];
                // relu(sign(ld) * du); ld==0 -> 0 handles equal labels
                float t = (ld > 0.0f) ? dd : ((ld < 0.0f) ? -dd : 0.0f);
                acc += fmaxf(t, 0.0f);
            }
        }
    }

    // wave32 reduction
#pragma unroll
    for (int off = 16; off > 0; off >>= 1)
        acc += __shfl_down(acc, off, 32);
    if (lane == 0) partials[w] = acc;   // every launched wave writes (0 if idle)
}

// Kernel 3: deterministic fixed-order reduction of per-wave partials.
__global__ void mmrl_reduce_kernel(const float* __restrict__ partials, int n,
                                   float* __restrict__ out, double inv_denom) {
    __shared__ double sm[BLOCK_THREADS];
    double s = 0.0;
    for (int i = threadIdx.x; i < n; i += BLOCK_THREADS)
        s += (double)partials[i];
    sm[threadIdx.x] = s;
    __syncthreads();
    for (int stride = BLOCK_THREADS / 2; stride > 0; stride >>= 1) {
        if ((int)threadIdx.x < stride)
            sm[threadIdx.x] += sm[threadIdx.x + stride];
        __syncthreads();
    }
    if (threadIdx.x == 0)
        out[0] = (float)(sm[0] * inv_denom);
}

extern "C" void kernel_launch(void* const* d_in, const int* in_sizes, int n_in,
                              void* d_out, int out_size, void* d_ws, size_t ws_size,
                              hipStream_t stream) {
    const float* scores = (const float*)d_in[0];
    const int*   labels = (const int*)d_in[1];
    const int*   mask   = (const int*)d_in[2];

    const int M   = in_sizes[2];            // 8192
    const int M16 = (M + 15) & ~15;
    const int T   = M16 / TILE;             // tiles per dimension (512)
    const int nchunk = (T + CHUNK - 1) / CHUNK;
    const int nwork  = T * nchunk;          // waves doing real work (8192)
    const int blocks = (nwork + WAVES_PER_BLOCK - 1) / WAVES_PER_BLOCK;
    const int nwaves = blocks * WAVES_PER_BLOCK;

    float* g_l      = (float*)d_ws;         // M16 floats
    float* g_u      = g_l + M16;            // M16 floats
    float* partials = g_u + M16;            // nwaves floats  (~96 KB total)

    mmrl_gather_kernel<<<(M16 + 255) / 256, 256, 0, stream>>>(
        scores, labels, mask, g_l, g_u, M, M16);

    mmrl_pair_tile_kernel<<<blocks, BLOCK_THREADS, 0, stream>>>(
        g_l, g_u, partials, T, nchunk);

    const double denom = (double)M * (double)(M - 1);
    mmrl_reduce_kernel<<<1, BLOCK_THREADS, 0, stream>>>(
        partials, nwaves, (float*)d_out, 1.0 / denom);
}